// RelationalConv_67044439491170
// MI455X (gfx1250) — compile-verified
//
#include <hip/hip_runtime.h>
#include <hip/hip_bf16.h>
#include <stdint.h>

typedef __attribute__((ext_vector_type(2))) float v2f;
typedef __attribute__((ext_vector_type(4))) float v4f;
typedef __attribute__((ext_vector_type(8))) float v8f;
typedef __attribute__((ext_vector_type(4))) unsigned int u32x4;
typedef __attribute__((ext_vector_type(4))) int          i32x4;
typedef __attribute__((ext_vector_type(8))) int          i32x8;

#define IN_CH   128
#define OUT_CH  128
#define N_BASES 8

// ---------------------------------------------------------------------------
// K0: zero the degree array
// ---------------------------------------------------------------------------
__global__ __launch_bounds__(256) void rgcn_zero_f32(float* __restrict__ p, int n) {
    int i = blockIdx.x * blockDim.x + threadIdx.x;
    if (i < n) p[i] = 0.0f;
}

// ---------------------------------------------------------------------------
// K1: in-degree count (f32 atomics; exact up to 2^24 which covers E=1.6M)
// ---------------------------------------------------------------------------
__global__ __launch_bounds__(256) void rgcn_count_deg(const int* __restrict__ edges,
                                                      float* __restrict__ deg,
                                                      int n_edges) {
    int e = blockIdx.x * blockDim.x + threadIdx.x;
    if (e < n_edges) {
        int dst = edges[e * 3 + 2];
        atomicAdd(&deg[dst], 1.0f);
    }
}

// ---------------------------------------------------------------------------
// K2: WMMA GEMM with TDM-staged weights.
//   grid.x tiles nodes by 128 (8 waves x 16 rows)
//   grid.y in [0,8]: 0..7 -> Y[:,b,:] = x @ basis[b]; 8 -> out = x@root_w + b
//
// The 128x128 f32 weight matrix (64 KB) is DMA'd into LDS once per block by
// the Tensor Data Mover (tensor_load_to_lds, TENSORcnt), then all 8 waves
// feed their WMMA B operands from LDS.  A operands stream from global (rows
// are unique per wave).  Full f32 WMMA (V_WMMA_F32_16X16X4_F32): the GEMM is
// compute-trivial vs. the memory-bound edge phase, so keep exact precision.
//
// f32 WMMA operand layouts (ISA 7.12.2):
//   A (16x4): lanes 0-15: v[0]=K0,v[1]=K1 (M=lane);  lanes16-31: v[0]=K2,v[1]=K3
//   B (4x16): lanes 0-15: v[0]=row K0,v[1]=row K1 (N=lane); lanes16-31: K2,K3
//   C/D (16x16): VGPR v -> M = v (lanes 0-15) / v+8 (lanes 16-31), N = lane&15
// ---------------------------------------------------------------------------
__global__ __launch_bounds__(256) void rgcn_gemm_wmma(
        const float* __restrict__ x,       // (N,128)
        const float* __restrict__ basis,   // (8,128,128)
        const float* __restrict__ root_w,  // (128,128)
        const float* __restrict__ root_b,  // (128)
        float* __restrict__ Y,             // ws (N,8,128)
        float* __restrict__ out,           // (N,128)
        int n_nodes) {
    __shared__ float shW[IN_CH * OUT_CH];                 // 64 KB of the 320 KB/WGP

    const int lane = threadIdx.x & 31;
    const int wave = threadIdx.x >> 5;                    // 0..7
    const int m0   = blockIdx.x * 128 + wave * 16;        // first row of this wave
    const int bsel = blockIdx.y;                          // 0..8
    const float* W = (bsel < N_BASES) ? (basis + (size_t)bsel * IN_CH * OUT_CH)
                                      : root_w;

#if __has_builtin(__builtin_amdgcn_tensor_load_to_lds)
    // --- TDM: one descriptor, whole 128x128 f32 tile -> LDS (wave 0 only) ---
    if (wave == 0) {
        const unsigned lds_off = (unsigned)(uintptr_t)(&shW[0]);   // low 32 bits = LDS offset
        const unsigned long long ga = (unsigned long long)(uintptr_t)W;

        u32x4 g0;
        g0.x = 1u;                                        // count=1 (valid), is_restore=0
        g0.y = lds_off;                                   // lds_addr [63:32]
        g0.z = (unsigned)(ga & 0xFFFFFFFFu);              // global_addr [95:64]
        g0.w = (unsigned)((ga >> 32) & 0x01FFFFFFu)       // global_addr [120:96]
             | (2u << 30);                                // type=2 ("image")

        i32x8 g1;
        g1[0] = (int)(2u << 16);                          // data_size=2 (4 bytes); wg_mask=0
        g1[1] = (int)((unsigned)IN_CH << 16);             // tensor_dim0[15:0] @ bits 63:48
        g1[2] = (int)((unsigned)IN_CH << 16);             // dim0 hi=0; tensor_dim1[15:0] @ 95:80
        g1[3] = (int)((unsigned)OUT_CH << 16);            // dim1 hi=0; tile_dim0 @ 127:112
        g1[4] = (int)(unsigned)IN_CH;                     // tile_dim1 @ 143:128; tile_dim2=0
        g1[5] = (int)(unsigned)OUT_CH;                    // tensor_dim0_stride[31:0]
        g1[6] = (int)((unsigned)(IN_CH * OUT_CH) << 16);  // stride hi=0; dim1_stride[15:0]
        g1[7] = 0;                                        // dim1_stride hi

        i32x4 gz4 = {0, 0, 0, 0};
        i32x8 gz8 = {0, 0, 0, 0, 0, 0, 0, 0};
        // 6-arg variant (clang-23 / therock-10.0 headers)
        __builtin_amdgcn_tensor_load_to_lds(g0, g1, gz4, gz4, gz8, 0);
        __builtin_amdgcn_s_wait_tensorcnt(0);
    }
#else
    // Fallback: cooperative copy of W into LDS
    for (int i = threadIdx.x; i < IN_CH * OUT_CH; i += 256) shW[i] = W[i];
#endif
    __syncthreads();

    const int nCol  = lane & 15;                          // N within 16-tile
    const int kHalf = (lane >> 4) << 1;                   // 0 (lanes<16) or 2

    // A source row for this lane, clamped so OOB rows still load valid memory
    // (no early exit: all waves reach the barrier; EXEC all-1s through WMMA).
    int rowA = m0 + (lane & 15);
    if (rowA >= n_nodes) rowA = n_nodes - 1;
    if (rowA < 0) rowA = 0;
    const float* xrow = x + (size_t)rowA * IN_CH + kHalf; // 8B-aligned

    v8f acc[8] = {};   // 8 N-tiles of 16 -> 16 x 128 strip per wave

    for (int k0 = 0; k0 < IN_CH; k0 += 4) {
        v2f a = *(const v2f*)(xrow + k0);                 // K = k0+kHalf, +1
        const float* Wk = &shW[(k0 + kHalf) * OUT_CH + nCol];
#pragma unroll
        for (int nt = 0; nt < 8; ++nt) {
            v2f b;
            b.x = Wk[nt * 16];                            // LDS: row K=k0+kHalf
            b.y = Wk[nt * 16 + OUT_CH];                   // LDS: row K=k0+kHalf+1
            acc[nt] = __builtin_amdgcn_wmma_f32_16x16x4_f32(
                /*neg_a=*/false, a, /*neg_b=*/false, b,
                /*c_mod=*/(short)0, acc[nt],
                /*reuse_a=*/false, /*reuse_b=*/false);
        }
    }

    const int rOff = (lane < 16) ? 0 : 8;
    if (bsel < N_BASES) {
#pragma unroll
        for (int nt = 0; nt < 8; ++nt) {
#pragma unroll
            for (int v = 0; v < 8; ++v) {
                int r = m0 + rOff + v;
                if (r < n_nodes)
                    Y[((size_t)r * N_BASES + bsel) * OUT_CH + nt * 16 + nCol] = acc[nt][v];
            }
        }
    } else {
#pragma unroll
        for (int nt = 0; nt < 8; ++nt) {
            float bias = root_b[nt * 16 + nCol];
#pragma unroll
            for (int v = 0; v < 8; ++v) {
                int r = m0 + rOff + v;
                if (r < n_nodes)
                    out[(size_t)r * OUT_CH + nt * 16 + nCol] = acc[nt][v] + bias;
            }
        }
    }
}

// ---------------------------------------------------------------------------
// K3: per-edge message + scatter.  One wave per edge; each lane owns 4
//     channels (float4).  Gathers the 8 basis rows of Y[src] (4KB/edge,
//     mostly L2 hits), combines with att[rel], scales by 1/deg[dst],
//     atomic-adds into out[dst].  This is the roofline-dominant phase
//     (~6.6 GB gather + ~0.8 GB atomics at 23.3 TB/s).
// ---------------------------------------------------------------------------
__global__ __launch_bounds__(256) void rgcn_msg(
        const int*   __restrict__ edges,  // (E,3)
        const float* __restrict__ att,    // (64,8)
        const float* __restrict__ Y,      // (N,8,128)
        const float* __restrict__ deg,    // (N)
        float* __restrict__ out,          // (N,128)
        int n_edges) {
    const int lane = threadIdx.x & 31;
    const int wave = threadIdx.x >> 5;
    const int e = blockIdx.x * 8 + wave;
    if (e >= n_edges) return;

    const int src = edges[e * 3 + 0];
    const int rel = edges[e * 3 + 1];
    const int dst = edges[e * 3 + 2];

    float d    = deg[dst];
    float dinv = (d > 0.0f) ? (1.0f / d) : 0.0f;

    const float* arow = att + rel * N_BASES;
    const float* ys   = Y + (size_t)src * (N_BASES * OUT_CH) + lane * 4;

    v4f m = {0.0f, 0.0f, 0.0f, 0.0f};
#pragma unroll
    for (int b = 0; b < N_BASES; ++b) {
        float ab = arow[b];
        v4f yb = *(const v4f*)(ys + b * OUT_CH);
        m += ab * yb;
    }
    m *= dinv;

    float* op = out + (size_t)dst * OUT_CH + lane * 4;
    atomicAdd(op + 0, m.x);
    atomicAdd(op + 1, m.y);
    atomicAdd(op + 2, m.z);
    atomicAdd(op + 3, m.w);
}

// ---------------------------------------------------------------------------
extern "C" void kernel_launch(void* const* d_in, const int* in_sizes, int n_in,
                              void* d_out, int out_size, void* d_ws, size_t ws_size,
                              hipStream_t stream) {
    const float* x      = (const float*)d_in[0];
    const int*   edges  = (const int*)  d_in[1];
    // d_in[2] = num_nodes (device scalar) — derive on host from in_sizes instead
    const float* basis  = (const float*)d_in[3];
    const float* att    = (const float*)d_in[4];
    const float* root_w = (const float*)d_in[5];
    const float* root_b = (const float*)d_in[6];
    float*       out    = (float*)d_out;

    const int n_nodes = in_sizes[0] / IN_CH;
    const int n_edges = in_sizes[1] / 3;

    // workspace layout: deg (n_nodes f32, 256B-aligned pad) | Y (n_nodes*8*128 f32)
    float* deg = (float*)d_ws;
    size_t degBytes = ((size_t)n_nodes * sizeof(float) + 255) & ~(size_t)255;
    float* Y = (float*)((char*)d_ws + degBytes);

    // K0: zero degrees
    rgcn_zero_f32<<<(n_nodes + 255) / 256, 256, 0, stream>>>(deg, n_nodes);
    // K1: count in-degrees
    rgcn_count_deg<<<(n_edges + 255) / 256, 256, 0, stream>>>(edges, deg, n_edges);
    // K2: WMMA GEMMs — Y[:,b,:] for b=0..7 and out = x@root_w + root_b
    dim3 gGemm((n_nodes + 127) / 128, N_BASES + 1);
    rgcn_gemm_wmma<<<gGemm, 256, 0, stream>>>(x, basis, root_w, root_b, Y, out, n_nodes);
    // K3: edge messages, normalized scatter-add into out
    rgcn_msg<<<(n_edges + 7) / 8, 256, 0, stream>>>(edges, att, Y, deg, out, n_edges);
}